// GatedMoE_2396591751256
// MI455X (gfx1250) — compile-verified
//
#include <hip/hip_runtime.h>
#include <math.h>

// Problem dims (from reference): B=8, T=512, D=128, E=256
#define N_TOK    4096        // B*T
#define D_DIM    128
#define E_NUM    256
#define N_ASSIGN (N_TOK * 2) // top-2 assignments

typedef float v2f __attribute__((ext_vector_type(2)));
typedef float v8f __attribute__((ext_vector_type(8)));
typedef unsigned int u32x4 __attribute__((ext_vector_type(4)));
typedef int i32x4 __attribute__((ext_vector_type(4)));
typedef int i32x8 __attribute__((ext_vector_type(8)));

// Toolchain discrimination: only amdgpu-toolchain (clang-23 + therock headers)
// ships the TDM descriptor header, and it uses the 6-arg builtin; ROCm 7.2
// uses the 5-arg builtin.
#if __has_include(<hip/amd_detail/amd_gfx1250_TDM.h>)
  #define TDM_6ARG 1
#else
  #define TDM_6ARG 0
#endif
#if __has_builtin(__builtin_amdgcn_tensor_load_to_lds) && \
    __has_builtin(__builtin_amdgcn_s_wait_tensorcnt)
  #define USE_TDM 1
#else
  #define USE_TDM 0
#endif

struct RoutRec { int e0, e1; float w0, w1; };

// ---------------------------------------------------------------------------
// Kernel 1: router GEMM  logits[N_TOK,E] = x[N_TOK,D] @ Wg[D,E] + bg
// One wave per 16x16 tile, f32 WMMA, K stepped by 4.
// A-frag: lane(0-15) row m0+lm, K = k0 + 2*(lane>>4) + {0,1}
// B-frag: lane col n0+lm,        K = k0 + 2*(lane>>4) + {0,1}
// C: vgpr r, lane-half h -> row m0 + r + 8h, col n0+lm
// ---------------------------------------------------------------------------
__global__ void __launch_bounds__(256) k_router_gemm(
    const float* __restrict__ x, const float* __restrict__ Wg,
    const float* __restrict__ bg, float* __restrict__ logits) {
  const int wave = threadIdx.x >> 5, lane = threadIdx.x & 31;
  const int tile = blockIdx.x * 8 + wave;   // 4096 tiles = 256 x 16
  const int tm = tile >> 4, tn = tile & 15;
  const int m0 = tm * 16, n0 = tn * 16;
  const int half = lane >> 4, lm = lane & 15;

  const float* arow = x  + (size_t)(m0 + lm) * D_DIM + half * 2;
  const float* bcol = Wg + (size_t)(half * 2) * E_NUM + n0 + lm;
  v8f acc = {};
#pragma unroll
  for (int k0 = 0; k0 < D_DIM; k0 += 4) {
    v2f a; a.x = arow[k0];          a.y = arow[k0 + 1];
    v2f b; b.x = bcol[k0 * E_NUM];  b.y = bcol[k0 * E_NUM + E_NUM];
    acc = __builtin_amdgcn_wmma_f32_16x16x4_f32(false, a, false, b,
                                                (short)0, acc, false, false);
  }
  const float bias = bg[n0 + lm];
#pragma unroll
  for (int r = 0; r < 8; ++r)
    logits[(size_t)(m0 + r + half * 8) * E_NUM + n0 + lm] = acc[r] + bias;
}

// ---------------------------------------------------------------------------
// Kernel 2: zero expert counts
// ---------------------------------------------------------------------------
__global__ void k_init(int* __restrict__ counts) { counts[threadIdx.x] = 0; }

// ---------------------------------------------------------------------------
// Kernel 3: per-token softmax + top-2 + normalized weights (one wave/token)
// ---------------------------------------------------------------------------
__global__ void __launch_bounds__(256) k_topk(
    const float* __restrict__ logits, RoutRec* __restrict__ rout,
    int* __restrict__ counts) {
  const int wave = threadIdx.x >> 5, lane = threadIdx.x & 31;
  const int t = blockIdx.x * 8 + wave;
  const float* lg = logits + (size_t)t * E_NUM;

  float v[8];
#pragma unroll
  for (int j = 0; j < 8; ++j) v[j] = lg[j * 32 + lane];

  // argmax #1 (ties -> lowest index, matches lax.top_k)
  float m1 = -INFINITY; int i1 = 0x7fffffff;
#pragma unroll
  for (int j = 0; j < 8; ++j) {
    int idx = j * 32 + lane;
    if (v[j] > m1 || (v[j] == m1 && idx < i1)) { m1 = v[j]; i1 = idx; }
  }
  for (int off = 16; off > 0; off >>= 1) {
    float om = __shfl_xor(m1, off, 32); int oi = __shfl_xor(i1, off, 32);
    if (om > m1 || (om == m1 && oi < i1)) { m1 = om; i1 = oi; }
  }
  // argmax #2 (exclude i1)
  float m2 = -INFINITY; int i2 = 0x7fffffff;
#pragma unroll
  for (int j = 0; j < 8; ++j) {
    int idx = j * 32 + lane;
    if (idx != i1 && (v[j] > m2 || (v[j] == m2 && idx < i2))) { m2 = v[j]; i2 = idx; }
  }
  for (int off = 16; off > 0; off >>= 1) {
    float om = __shfl_xor(m2, off, 32); int oi = __shfl_xor(i2, off, 32);
    if (om > m2 || (om == m2 && oi < i2)) { m2 = om; i2 = oi; }
  }
  // softmax denominator
  float s = 0.f;
#pragma unroll
  for (int j = 0; j < 8; ++j) s += expf(v[j] - m1);
  for (int off = 16; off > 0; off >>= 1) s += __shfl_xor(s, off, 32);

  if (lane == 0) {
    float p1 = 1.0f / s;                 // exp(m1-m1)/Z
    float p2 = expf(m2 - m1) / s;
    float den = p1 + p2 + 1e-6f;
    RoutRec r; r.e0 = i1; r.e1 = i2; r.w0 = p1 / den; r.w1 = p2 / den;
    rout[t] = r;
    atomicAdd(&counts[i1], 1);
    atomicAdd(&counts[i2], 1);
  }
}

// ---------------------------------------------------------------------------
// Kernel 4: exclusive prefix scan of 256 counts (single block)
// ---------------------------------------------------------------------------
__global__ void k_scan(const int* __restrict__ counts,
                       int* __restrict__ offsets, int* __restrict__ cursor) {
  __shared__ int sh[E_NUM];
  int tid = threadIdx.x;
  int c = counts[tid];
  sh[tid] = c; __syncthreads();
  for (int d = 1; d < E_NUM; d <<= 1) {
    int add = (tid >= d) ? sh[tid - d] : 0;
    __syncthreads();
    sh[tid] += add;
    __syncthreads();
  }
  int excl = sh[tid] - c;
  offsets[tid] = excl;
  cursor[tid]  = excl;
}

// ---------------------------------------------------------------------------
// Kernel 5: scatter assignments into expert-sorted order + inverse map
// (atomic slot order varies run-to-run, but inv[] makes results invariant)
// ---------------------------------------------------------------------------
__global__ void k_fill(const RoutRec* __restrict__ rout, int* __restrict__ cursor,
                       int* __restrict__ assign, int* __restrict__ inv) {
  int i = blockIdx.x * blockDim.x + threadIdx.x;  // 0..8191
  int t = i >> 1, s = i & 1;
  RoutRec r = rout[t];
  int e = s ? r.e1 : r.e0;
  int pos = atomicAdd(&cursor[e], 1);
  assign[pos] = i;            // encodes (token<<1)|slot
  inv[i] = pos;
}

// ---------------------------------------------------------------------------
// Kernel 6: per-expert GEMM. One block per expert; the 64 KB expert matrix is
// staged into LDS by the Tensor Data Mover (single tensor_load_to_lds issued
// by wave 0, tracked by TENSORcnt), then 8 waves each own a 16-col strip and
// sweep token rows in chunks of 16 with f32 WMMA (B-frags served from LDS).
// ---------------------------------------------------------------------------
__global__ void __launch_bounds__(256) k_expert_gemm(
    const float* __restrict__ x, const float* __restrict__ expert_w,
    const int* __restrict__ offsets, const int* __restrict__ counts,
    const int* __restrict__ assign, float* __restrict__ y) {
  __shared__ float Wlds[D_DIM * D_DIM];   // 64 KB of the WGP's 320 KB
  const int e = blockIdx.x;
  const int tid = threadIdx.x;

#if USE_TDM
  if (tid < 32) {   // wave 0 issues the DMA; TDM ignores EXEC
    unsigned long long ga =
        (unsigned long long)(uintptr_t)(expert_w + (size_t)e * D_DIM * D_DIM);
    unsigned int lds_off = (unsigned int)(uintptr_t)(void*)Wlds;
    // D# group 0: count=1 | lds_addr | global_addr[56:0] | type=2
    u32x4 g0;
    g0[0] = 1u;
    g0[1] = lds_off;
    g0[2] = (unsigned int)(ga & 0xffffffffu);
    g0[3] = (unsigned int)((ga >> 32) & 0x01ffffffu) | (2u << 30);
    // D# group 1: data_size=2 (4B); tensor 128x128; tile 128x128; stride 128
    i32x8 g1;
    g1[0] = 0x00020000;          // [17:16] data_size = 2 -> 4-byte elements
    g1[1] = (int)(128u << 16);   // tensor_dim0[15:0] = 128   (bits 63:48)
    g1[2] = (int)(128u << 16);   // tensor_dim1[15:0] = 128   (bits 95:80)
    g1[3] = (int)(128u << 16);   // tile_dim0 = 128           (bits 127:112)
    g1[4] = 128;                 // tile_dim1 = 128, tile_dim2 = 0
    g1[5] = 128;                 // tensor_dim0_stride[31:0] = 128
    g1[6] = 0;
    g1[7] = 0;
    i32x4 gz = {};               // groups 2/3 unused (<=2D tensor)
  #if TDM_6ARG
    i32x8 gz8 = {};
    __builtin_amdgcn_tensor_load_to_lds(g0, g1, gz, gz, gz8, 0);
  #else
    __builtin_amdgcn_tensor_load_to_lds(g0, g1, gz, gz, 0);
  #endif
    __builtin_amdgcn_s_wait_tensorcnt(0);
  }
  __syncthreads();
#else
  const float4* src = (const float4*)(expert_w + (size_t)e * D_DIM * D_DIM);
  float4* dst = (float4*)Wlds;
  for (int i = tid; i < D_DIM * D_DIM / 4; i += 256) dst[i] = src[i];
  __syncthreads();
#endif

  const int start = offsets[e];
  const int n     = counts[e];
  const int wave = tid >> 5, lane = tid & 31;
  const int half = lane >> 4, lm = lane & 15;
  const int n0 = wave * 16;

  for (int chunk = 0; chunk < n; chunk += 16) {
    int rowA = chunk + lm;
    int tok = (rowA < n) ? (assign[start + rowA] >> 1) : 0;
    const float* arow = x + (size_t)tok * D_DIM + half * 2;
    const float* bcol = Wlds + (half * 2) * D_DIM + n0 + lm;
    v8f acc = {};
#pragma unroll
    for (int k0 = 0; k0 < D_DIM; k0 += 4) {
      v2f a; a.x = arow[k0];          a.y = arow[k0 + 1];
      v2f b; b.x = bcol[k0 * D_DIM];  b.y = bcol[k0 * D_DIM + D_DIM];
      acc = __builtin_amdgcn_wmma_f32_16x16x4_f32(false, a, false, b,
                                                  (short)0, acc, false, false);
    }
#pragma unroll
    for (int r = 0; r < 8; ++r) {
      int row = chunk + r + half * 8;
      if (row < n) y[(size_t)(start + row) * D_DIM + n0 + lm] = acc[r];
    }
  }
}

// ---------------------------------------------------------------------------
// Kernel 7: mixed[t] = w0*y0[t] + w1*y1[t]
// ---------------------------------------------------------------------------
__global__ void k_mix(const float* __restrict__ y, const RoutRec* __restrict__ rout,
                      const int* __restrict__ inv, float* __restrict__ mixed) {
  int i = blockIdx.x * blockDim.x + threadIdx.x;  // 0..524287
  int t = i >> 7, o = i & 127;
  RoutRec r = rout[t];
  float y0 = y[(size_t)inv[2 * t]     * D_DIM + o];
  float y1 = y[(size_t)inv[2 * t + 1] * D_DIM + o];
  mixed[i] = r.w0 * y0 + r.w1 * y1;
}

// ---------------------------------------------------------------------------
// Kernel 8: fused SwiGLU: G = mixed@W1+b1, H = mixed@W2+b2 (two WMMA
// accumulators per wave, shared A-frag), out = G*sigmoid(G)*H -> d_out
// ---------------------------------------------------------------------------
__global__ void __launch_bounds__(256) k_swiglu(
    const float* __restrict__ mixed,
    const float* __restrict__ W1, const float* __restrict__ b1,
    const float* __restrict__ W2, const float* __restrict__ b2,
    float* __restrict__ out) {
  const int wave = threadIdx.x >> 5, lane = threadIdx.x & 31;
  const int tile = blockIdx.x * 8 + wave;   // 2048 tiles = 256 x 8
  const int tm = tile >> 3, tn = tile & 7;
  const int m0 = tm * 16, n0 = tn * 16;
  const int half = lane >> 4, lm = lane & 15;

  const float* arow = mixed + (size_t)(m0 + lm) * D_DIM + half * 2;
  const float* c1 = W1 + (size_t)(half * 2) * D_DIM + n0 + lm;
  const float* c2 = W2 + (size_t)(half * 2) * D_DIM + n0 + lm;
  v8f accG = {}, accH = {};
#pragma unroll
  for (int k0 = 0; k0 < D_DIM; k0 += 4) {
    v2f a;  a.x  = arow[k0];         a.y  = arow[k0 + 1];
    v2f bG; bG.x = c1[k0 * D_DIM];   bG.y = c1[k0 * D_DIM + D_DIM];
    v2f bH; bH.x = c2[k0 * D_DIM];   bH.y = c2[k0 * D_DIM + D_DIM];
    accG = __builtin_amdgcn_wmma_f32_16x16x4_f32(false, a, false, bG,
                                                 (short)0, accG, false, false);
    accH = __builtin_amdgcn_wmma_f32_16x16x4_f32(false, a, false, bH,
                                                 (short)0, accH, false, false);
  }
  const float bias1 = b1[n0 + lm], bias2 = b2[n0 + lm];
#pragma unroll
  for (int r = 0; r < 8; ++r) {
    float g = accG[r] + bias1;
    float h = accH[r] + bias2;
    float sig = 1.0f / (1.0f + expf(-g));
    out[(size_t)(m0 + r + half * 8) * D_DIM + n0 + lm] = g * sig * h;
  }
}

// ---------------------------------------------------------------------------
// Kernel 9: consensus[t] = exp(-mean_o(w0*(y0-wa)^2 + w1*(y1-wa)^2))
// one wave per token
// ---------------------------------------------------------------------------
__global__ void __launch_bounds__(256) k_consensus(
    const float* __restrict__ wa_out, const float* __restrict__ y,
    const RoutRec* __restrict__ rout, const int* __restrict__ inv,
    float* __restrict__ cons) {
  const int wave = threadIdx.x >> 5, lane = threadIdx.x & 31;
  const int t = blockIdx.x * 8 + wave;
  RoutRec r = rout[t];
  const float* wa = wa_out + (size_t)t * D_DIM;
  const float* y0 = y + (size_t)inv[2 * t]     * D_DIM;
  const float* y1 = y + (size_t)inv[2 * t + 1] * D_DIM;
  float acc = 0.f;
#pragma unroll
  for (int j = 0; j < 4; ++j) {
    int o = j * 32 + lane;
    float w = wa[o];
    float d0 = y0[o] - w, d1 = y1[o] - w;
    acc += r.w0 * d0 * d0 + r.w1 * d1 * d1;
  }
  for (int off = 16; off > 0; off >>= 1) acc += __shfl_xor(acc, off, 32);
  if (lane == 0) cons[t] = expf(-acc * (1.0f / D_DIM));
}

// ---------------------------------------------------------------------------
extern "C" void kernel_launch(void* const* d_in, const int* in_sizes, int n_in,
                              void* d_out, int out_size, void* d_ws, size_t ws_size,
                              hipStream_t stream) {
  const float* x  = (const float*)d_in[0];   // [4096,128]
  const float* Wg = (const float*)d_in[1];   // [128,256]
  const float* bg = (const float*)d_in[2];   // [256]
  const float* ew = (const float*)d_in[3];   // [256,128,128]
  const float* W1 = (const float*)d_in[4];   // [128,128]
  const float* b1 = (const float*)d_in[5];   // [128]
  const float* W2 = (const float*)d_in[6];   // [128,128]
  const float* b2 = (const float*)d_in[7];   // [128]
  float* out = (float*)d_out;                // wa [524288] then consensus [4096]

  // workspace carve-up (~10.6 MB total)
  char* ws = (char*)d_ws;
  float*   logits  = (float*)ws;    ws += (size_t)N_TOK * E_NUM * sizeof(float);
  float*   y       = (float*)ws;    ws += (size_t)N_ASSIGN * D_DIM * sizeof(float);
  float*   mixed   = (float*)ws;    ws += (size_t)N_TOK * D_DIM * sizeof(float);
  RoutRec* rout    = (RoutRec*)ws;  ws += (size_t)N_TOK * sizeof(RoutRec);
  int*     assign  = (int*)ws;      ws += (size_t)N_ASSIGN * sizeof(int);
  int*     inv     = (int*)ws;      ws += (size_t)N_ASSIGN * sizeof(int);
  int*     counts  = (int*)ws;      ws += (size_t)E_NUM * sizeof(int);
  int*     offsets = (int*)ws;      ws += (size_t)E_NUM * sizeof(int);
  int*     cursor  = (int*)ws;      ws += (size_t)E_NUM * sizeof(int);

  k_init       <<<1,    256, 0, stream>>>(counts);
  k_router_gemm<<<512,  256, 0, stream>>>(x, Wg, bg, logits);
  k_topk       <<<512,  256, 0, stream>>>(logits, rout, counts);
  k_scan       <<<1,    256, 0, stream>>>(counts, offsets, cursor);
  k_fill       <<<32,   256, 0, stream>>>(rout, cursor, assign, inv);
  k_expert_gemm<<<256,  256, 0, stream>>>(x, ew, offsets, counts, assign, y);
  k_mix        <<<2048, 256, 0, stream>>>(y, rout, inv, mixed);
  k_swiglu     <<<256,  256, 0, stream>>>(mixed, W1, b1, W2, b2, out);
  k_consensus  <<<512,  256, 0, stream>>>(out, y, rout, inv, out + (size_t)N_TOK * D_DIM);
}